// GCN_18854906429732
// MI455X (gfx1250) — compile-verified
//
#include <hip/hip_runtime.h>
#include <hip/hip_bf16.h>

// GCN forward, MI455X (gfx1250, wave32).
// Memory-bound workload -> keep full f32 numerics and use V_WMMA_F32_16X16X4_F32
// for both dense GEMMs. Scatter phases use global f32 atomics (L2-resident).

typedef __attribute__((ext_vector_type(2))) float v2f;
typedef __attribute__((ext_vector_type(8))) float v8f;

#define N_NODES 50000
#define N_EDGES 800000
#define NFEAT   512
#define NHID    128
#define NCLASS  40
#define NC_PAD  48   // 3 col-tiles of 16

// ---------------------------------------------------------------------------
// init agg1[i,f] = b1[f]
__global__ void gcn_init_agg1(float* __restrict__ agg1, const float* __restrict__ b1) {
    int idx = blockIdx.x * blockDim.x + threadIdx.x;
    if (idx < N_NODES * NHID) {
        agg1[idx] = b1[idx & (NHID - 1)];
    }
}

// init out[i,c] = b2[c]
__global__ void gcn_init_out(float* __restrict__ out, const float* __restrict__ b2) {
    int idx = blockIdx.x * blockDim.x + threadIdx.x;
    if (idx < N_NODES * NCLASS) {
        out[idx] = b2[idx % NCLASS];
    }
}

// ---------------------------------------------------------------------------
// GEMM1: support1[50000,128] = x[50000,512] @ W1[512,128]
// Block = 256 threads = 8 waves. Block owns 16 rows; wave w owns cols [16w,16w+16).
// A tile (16x512 f32 = 32 KB) staged in LDS, shared by all 8 waves.
__global__ void __launch_bounds__(256)
gcn_gemm1(const float* __restrict__ x, const float* __restrict__ W1,
          float* __restrict__ out) {
    __shared__ float As[16 * NFEAT];   // 32 KB

    const int row0 = blockIdx.x * 16;
    const int tid  = threadIdx.x;

    // cooperative stage of A tile (float4, 2048 vec loads / 256 threads)
    {
        const float4* xv  = (const float4*)(x + (size_t)row0 * NFEAT);
        float4*       av  = (float4*)As;
        #pragma unroll
        for (int i = 0; i < (16 * NFEAT / 4) / 256; ++i)
            av[tid + i * 256] = xv[tid + i * 256];
    }
    __syncthreads();

    const int wave = tid >> 5;
    const int lane = tid & 31;
    const int m    = lane & 15;     // A row within tile / D col selector
    const int half = lane >> 4;     // K-half for A/B, M-half for C/D
    const int nb   = wave * 16;     // column tile base
    const int n    = nb + (lane & 15);

    v8f acc = {};
    #pragma unroll 4
    for (int k0 = 0; k0 < NFEAT; k0 += 4) {
        const int kk = k0 + 2 * half;
        // A 16x4 f32 fragment: lane m, VGPR v -> A[m][k0 + 2*half + v]
        v2f a = *(const v2f*)(&As[m * NFEAT + kk]);
        // B 4x16 f32 fragment (mirror layout): lane n, VGPR v -> B[kk+v][n]
        v2f b;
        b.x = W1[(size_t)kk * NHID + n];
        b.y = W1[(size_t)(kk + 1) * NHID + n];
        acc = __builtin_amdgcn_wmma_f32_16x16x4_f32(
            /*neg_a=*/false, a, /*neg_b=*/false, b,
            /*c_mod=*/(short)0, acc, /*reuse_a=*/false, /*reuse_b=*/false);
    }

    // C/D layout: VGPR v, lanes 0-15 -> M=v, lanes 16-31 -> M=8+v; N = lane&15
    float* orow = out + (size_t)row0 * NHID + n;
    #pragma unroll
    for (int i = 0; i < 8; ++i)
        orow[(size_t)(8 * half + i) * NHID] = acc[i];
}

// ---------------------------------------------------------------------------
// GEMM2: support2[50000,40] = h[50000,128] @ W2[128,40]   (N padded to 48)
// Block = 96 threads = 3 waves, one 16-col tile per wave.
__global__ void __launch_bounds__(96)
gcn_gemm2(const float* __restrict__ h, const float* __restrict__ W2,
          float* __restrict__ out) {
    __shared__ float Hs[16 * NHID];    // 8 KB

    const int row0 = blockIdx.x * 16;
    const int tid  = threadIdx.x;

    {
        const float4* hv = (const float4*)(h + (size_t)row0 * NHID);
        float4*       sv = (float4*)Hs;
        for (int i = tid; i < 16 * NHID / 4; i += 96)
            sv[i] = hv[i];
    }
    __syncthreads();

    const int wave = tid >> 5;
    const int lane = tid & 31;
    const int m    = lane & 15;
    const int half = lane >> 4;
    const int n    = wave * 16 + (lane & 15);   // 0..47, valid if < 40

    const int   nc   = (n < NCLASS) ? n : (NCLASS - 1);  // clamp: keep loads in-bounds
    const float mask = (n < NCLASS) ? 1.0f : 0.0f;       // branchless -> EXEC all-1s

    v8f acc = {};
    #pragma unroll 4
    for (int k0 = 0; k0 < NHID; k0 += 4) {
        const int kk = k0 + 2 * half;
        v2f a = *(const v2f*)(&Hs[m * NHID + kk]);
        v2f b;
        b.x = W2[(size_t)kk * NCLASS + nc] * mask;
        b.y = W2[(size_t)(kk + 1) * NCLASS + nc] * mask;
        acc = __builtin_amdgcn_wmma_f32_16x16x4_f32(
            false, a, false, b, (short)0, acc, false, false);
    }

    if (n < NCLASS) {
        float* orow = out + (size_t)row0 * NCLASS + n;
        #pragma unroll
        for (int i = 0; i < 8; ++i)
            orow[(size_t)(8 * half + i) * NCLASS] = acc[i];
    }
}

// ---------------------------------------------------------------------------
// scatter1: one wave per edge; 32 lanes x float4 = 128 features.
// agg1[dst,:] += support1[src,:] * w   via global f32 atomics (L2-resident).
__global__ void __launch_bounds__(256)
gcn_scatter1(const float* __restrict__ support1,
             const int* __restrict__ esrc, const int* __restrict__ edst,
             const float* __restrict__ ew, float* __restrict__ agg1) {
    int gid = blockIdx.x * blockDim.x + threadIdx.x;
    int e   = gid >> 5;
    int j   = gid & 31;          // float4 slot: feats 4j..4j+3
    if (e >= N_EDGES) return;

    const int   src = esrc[e];
    const int   dst = edst[e];
    const float w   = ew[e];

    const float4 v = ((const float4*)(support1 + (size_t)src * NHID))[j];
    float* o = agg1 + (size_t)dst * NHID + 4 * j;
    atomicAdd(o + 0, v.x * w);
    atomicAdd(o + 1, v.y * w);
    atomicAdd(o + 2, v.z * w);
    atomicAdd(o + 3, v.w * w);
}

// relu in place on agg1 (float4)
__global__ void gcn_relu(float4* __restrict__ a) {
    int idx = blockIdx.x * blockDim.x + threadIdx.x;
    if (idx < N_NODES * NHID / 4) {
        float4 v = a[idx];
        v.x = fmaxf(v.x, 0.0f);
        v.y = fmaxf(v.y, 0.0f);
        v.z = fmaxf(v.z, 0.0f);
        v.w = fmaxf(v.w, 0.0f);
        a[idx] = v;
    }
}

// scatter2: 8 threads per edge, 5 features each (8*5 = 40).
__global__ void __launch_bounds__(256)
gcn_scatter2(const float* __restrict__ support2,
             const int* __restrict__ esrc, const int* __restrict__ edst,
             const float* __restrict__ ew, float* __restrict__ out) {
    int gid = blockIdx.x * blockDim.x + threadIdx.x;
    int e   = gid >> 3;
    int sub = gid & 7;
    if (e >= N_EDGES) return;

    const int   src = esrc[e];
    const int   dst = edst[e];
    const float w   = ew[e];

    const float* sp = support2 + (size_t)src * NCLASS + sub * 5;
    float*       op = out      + (size_t)dst * NCLASS + sub * 5;
    #pragma unroll
    for (int i = 0; i < 5; ++i)
        atomicAdd(op + i, sp[i] * w);
}

// ---------------------------------------------------------------------------
extern "C" void kernel_launch(void* const* d_in, const int* in_sizes, int n_in,
                              void* d_out, int out_size, void* d_ws, size_t ws_size,
                              hipStream_t stream) {
    const float* x    = (const float*)d_in[0];
    const int*   esrc = (const int*)  d_in[1];
    const int*   edst = (const int*)  d_in[2];
    const float* ew   = (const float*)d_in[3];
    const float* W1   = (const float*)d_in[4];
    const float* b1   = (const float*)d_in[5];
    const float* W2   = (const float*)d_in[6];
    const float* b2   = (const float*)d_in[7];
    float* out = (float*)d_out;

    // workspace layout (all 256B aligned):
    //   support1 : 50000*128*4 = 25,600,000 B
    //   agg1/h   : 25,600,000 B
    //   support2 : 50000*40*4  =  8,000,000 B    -> total 59.2 MB
    char*  ws       = (char*)d_ws;
    float* support1 = (float*)(ws);
    float* agg1     = (float*)(ws + 25600000);
    float* support2 = (float*)(ws + 51200000);

    // 1. agg1 = b1 (broadcast)
    gcn_init_agg1<<<(N_NODES * NHID + 255) / 256, 256, 0, stream>>>(agg1, b1);

    // 2. support1 = x @ W1   (f32 WMMA)
    gcn_gemm1<<<N_NODES / 16, 256, 0, stream>>>(x, W1, support1);

    // 3. agg1 += scatter(support1[src] * w -> dst)
    gcn_scatter1<<<(N_EDGES * 32) / 256, 256, 0, stream>>>(support1, esrc, edst, ew, agg1);

    // 4. h = relu(agg1) in place
    gcn_relu<<<(N_NODES * NHID / 4 + 255) / 256, 256, 0, stream>>>((float4*)agg1);

    // 5. support2 = h @ W2   (f32 WMMA, N padded 40->48)
    gcn_gemm2<<<N_NODES / 16, 96, 0, stream>>>(agg1, W2, support2);

    // 6. out = b2 (broadcast)
    gcn_init_out<<<(N_NODES * NCLASS + 255) / 256, 256, 0, stream>>>(out, b2);

    // 7. out += scatter(support2[src] * w -> dst)
    gcn_scatter2<<<(N_EDGES * 8) / 256, 256, 0, stream>>>(support2, esrc, edst, ew, out);
}